// MVRenderer_24326694764980
// MI455X (gfx1250) — compile-verified
//
#include <hip/hip_runtime.h>
#include <stdint.h>

// ---------------------------------------------------------------------------
// SoftRas-style MV renderer composite for MI455X (gfx1250).
// Bandwidth-bound streaming kernel (~436 MB @ 23.3 TB/s => ~19 us floor).
// CDNA5 path: TDM (tensor_load_to_lds) DMA-stages each block's 48 KB input
// tile into LDS (4 x 1-D descriptors, one per wave, s_wait_tensorcnt);
// compute reads LDS via b128; RGBA written with non-temporal B128 stores.
// ---------------------------------------------------------------------------

typedef __attribute__((ext_vector_type(4))) float        f32x4;
typedef __attribute__((ext_vector_type(4))) unsigned int u32x4;
typedef __attribute__((ext_vector_type(8))) int          i32x8;
typedef __attribute__((ext_vector_type(4))) int          i32x4;

#define K_FACES   8
#define C_FEAT    3
#define BLOCK     256

#define SIGMA_INV 1.0e4f          // 1/1e-4
#define GAMMA_INV 1.0e4f          // 1/1e-4
#define ZFARF     100.0f
#define ZNEARF    1.0f
#define INV_RANGE (1.0f / (ZFARF - ZNEARF))
#define EPSF      1e-10f

#ifndef __has_builtin
#define __has_builtin(x) 0
#endif

#if defined(__HIP_DEVICE_COMPILE__) && __has_builtin(__builtin_amdgcn_tensor_load_to_lds)
#define USE_TDM 1
#else
#define USE_TDM 0
#endif

#if USE_TDM
// 1-D TDM copy: tile of `tile_dwords` 32-bit elements starting at gsrc -> LDS
// at byte offset lds_byte_off. tensor_dim0 = remaining valid elements so the
// hardware zero-fills any tail overrun (OOB reads return zero).
__device__ __forceinline__ void tdm_copy_1d(uint32_t lds_byte_off,
                                            const void* gsrc,
                                            uint32_t tile_dwords,
                                            uint64_t tensor_dwords) {
  uint64_t ga = (uint64_t)(uintptr_t)gsrc;
  uint32_t td0 = (tensor_dwords > 0xFFFFFFFFull) ? 0xFFFFFFFFu
                                                 : (uint32_t)tensor_dwords;
  // D# group 0 (128b): count=1 | lds_addr | global_addr[56:0] | type=2
  u32x4 g0;
  g0[0] = 1u;                                       // count=1, user descriptor
  g0[1] = lds_byte_off;                             // lds_addr (bytes)
  g0[2] = (uint32_t)ga;                             // global_addr[31:0]
  g0[3] = (uint32_t)((ga >> 32) & 0x01FFFFFFu)      // global_addr[56:32]
        | (2u << 30);                               // type = 2 ("image")
  // D# group 1 (256b): data_size=4B, tensor_dim0=td0, tile_dim0=tile_dwords,
  // tile_dim1/2 unused, strides irrelevant for 1-D.
  i32x8 g1;
  g1[0] = (int)(2u << 16);                          // wg_mask=0, data_size=2 (4B)
  g1[1] = (int)((td0 & 0xFFFFu) << 16);             // tensor_dim0[15:0] @ bits 63:48
  g1[2] = (int)((td0 >> 16) & 0xFFFFu);             // tensor_dim0[31:16]; tensor_dim1=0
  g1[3] = (int)((tile_dwords & 0xFFFFu) << 16);     // tile_dim0 @ bits 127:112
  g1[4] = 0;                                        // tile_dim1=0, tile_dim2=0
  g1[5] = (int)td0;                                 // tensor_dim0_stride[31:0] (unused 1-D)
  g1[6] = 0;
  g1[7] = 0;
  i32x4 gz4 = {0, 0, 0, 0};                         // groups 2/3 unused (<=2-D)
  i32x8 gz8 = {0, 0, 0, 0, 0, 0, 0, 0};             // extra group (6-arg toolchain form)
  __builtin_amdgcn_tensor_load_to_lds(g0, g1, gz4, gz4, gz8, /*cpol=*/0);
}
#endif

__global__ void __launch_bounds__(BLOCK) mvrenderer_softras_kernel(
    const float* __restrict__ feat,   // (npix, K, C)
    const float* __restrict__ dists,  // (npix, K)
    const float* __restrict__ zbuf,   // (npix, K)
    const int*   __restrict__ p2f,    // (npix, K)
    float* __restrict__ out,          // (npix, C+1)
    int npix) {
  __shared__ int   s_p2f[BLOCK * K_FACES];            // 8 KB
  __shared__ float s_dst[BLOCK * K_FACES];            // 8 KB
  __shared__ float s_zbf[BLOCK * K_FACES];            // 8 KB
  __shared__ float s_ft [BLOCK * K_FACES * C_FEAT];   // 24 KB

  const int t    = threadIdx.x;
  const int pix0 = blockIdx.x * BLOCK;

#if USE_TDM
  {
    const int wave = t >> 5;                          // wave32
    if (wave < 4) {
      const uint64_t remK  = (uint64_t)(npix - pix0) * (uint64_t)K_FACES;
      const uint64_t remKC = remK * (uint64_t)C_FEAT;
      if (wave == 0)
        tdm_copy_1d((uint32_t)(uintptr_t)s_p2f, p2f   + (size_t)pix0 * K_FACES,
                    BLOCK * K_FACES, remK);
      if (wave == 1)
        tdm_copy_1d((uint32_t)(uintptr_t)s_dst, dists + (size_t)pix0 * K_FACES,
                    BLOCK * K_FACES, remK);
      if (wave == 2)
        tdm_copy_1d((uint32_t)(uintptr_t)s_zbf, zbuf  + (size_t)pix0 * K_FACES,
                    BLOCK * K_FACES, remK);
      if (wave == 3)
        tdm_copy_1d((uint32_t)(uintptr_t)s_ft,  feat  + (size_t)pix0 * K_FACES * C_FEAT,
                    BLOCK * K_FACES * C_FEAT, remKC);
      __builtin_amdgcn_s_wait_tensorcnt(0);           // each wave waits on its own TENSORcnt
    }
  }
#else
  {
    const int p = pix0 + t;
    if (p < npix) {
#pragma unroll
      for (int k = 0; k < K_FACES; ++k) {
        s_p2f[t * K_FACES + k] = p2f  [(size_t)p * K_FACES + k];
        s_dst[t * K_FACES + k] = dists[(size_t)p * K_FACES + k];
        s_zbf[t * K_FACES + k] = zbuf [(size_t)p * K_FACES + k];
      }
#pragma unroll
      for (int j = 0; j < K_FACES * C_FEAT; ++j)
        s_ft[t * K_FACES * C_FEAT + j] = feat[(size_t)p * K_FACES * C_FEAT + j];
    }
  }
#endif
  __syncthreads();
  // The TDM writes to LDS are invisible to the IR; escape the LDS addresses
  // through an opaque asm with a memory clobber so the optimizer cannot fold
  // the "never-stored" shared-memory loads away.
  asm volatile("" ::
               "s"((uint32_t)(uintptr_t)s_p2f), "s"((uint32_t)(uintptr_t)s_dst),
               "s"((uint32_t)(uintptr_t)s_zbf), "s"((uint32_t)(uintptr_t)s_ft)
               : "memory");

  const int p = pix0 + t;
  if (p < npix) {
    // Vectorized LDS reads: 12 x ds_load_b128 per thread (all 16B-aligned).
    i32x4 pv[2];
    f32x4 dv[2], zv[2], fv[6];
#pragma unroll
    for (int j = 0; j < 2; ++j) {
      pv[j] = ((const i32x4*)(s_p2f + t * K_FACES))[j];
      dv[j] = ((const f32x4*)(s_dst + t * K_FACES))[j];
      zv[j] = ((const f32x4*)(s_zbf + t * K_FACES))[j];
    }
#pragma unroll
    for (int j = 0; j < 6; ++j)
      fv[j] = ((const f32x4*)(s_ft + t * K_FACES * C_FEAT))[j];

    float prob[K_FACES], zinv[K_FACES];
    float alpha = 1.0f;
    float zmax  = 0.0f;
#pragma unroll
    for (int k = 0; k < K_FACES; ++k) {
      const bool  m  = pv[k >> 2][k & 3] >= 0;
      float       pr = 1.0f / (1.0f + __expf(dv[k >> 2][k & 3] * SIGMA_INV));
      pr             = m ? pr : 0.0f;                 // sigmoid(-d/sigma)*mask
      prob[k]        = pr;
      alpha         *= (1.0f - pr);
      const float zi = m ? (ZFARF - zv[k >> 2][k & 3]) * INV_RANGE : 0.0f;
      zinv[k]        = zi;
      zmax           = fmaxf(zmax, zi);
    }
    zmax = fmaxf(zmax, EPSF);                          // clip(max(z_inv), EPS)
    float denom = fmaxf(__expf((EPSF - zmax) * GAMMA_INV), EPSF);  // delta
    float r = 0.0f, g = 0.0f, b = 0.0f;
#pragma unroll
    for (int k = 0; k < K_FACES; ++k) {
      const float w = prob[k] * __expf((zinv[k] - zmax) * GAMMA_INV);
      denom += w;
      r = fmaf(w, fv[(k * 3 + 0) >> 2][(k * 3 + 0) & 3], r);
      g = fmaf(w, fv[(k * 3 + 1) >> 2][(k * 3 + 1) & 3], g);
      b = fmaf(w, fv[(k * 3 + 2) >> 2][(k * 3 + 2) & 3], b);
    }
    const float inv = 1.0f / denom;
    f32x4 o;
    o.x = r * inv;
    o.y = g * inv;
    o.z = b * inv;
    o.w = 1.0f - alpha;
    // Streaming output (436 MB total >> 192 MB L2): non-temporal B128 store.
    __builtin_nontemporal_store(o, (f32x4*)(out + (size_t)p * 4));
  }
}

extern "C" void kernel_launch(void* const* d_in, const int* in_sizes, int n_in,
                              void* d_out, int out_size, void* d_ws, size_t ws_size,
                              hipStream_t stream) {
  (void)n_in; (void)out_size; (void)d_ws; (void)ws_size;
  const float* feat  = (const float*)d_in[0];  // (N,H,W,K,C) f32
  const float* dists = (const float*)d_in[1];  // (N,H,W,K)   f32
  const float* zbuf  = (const float*)d_in[2];  // (N,H,W,K)   f32
  const int*   p2f   = (const int*)  d_in[3];  // (N,H,W,K)   i32
  float*       out   = (float*)d_out;          // (N,H,W,C+1) f32

  const int npix   = in_sizes[1] / K_FACES;    // N*H*W
  const int blocks = (npix + BLOCK - 1) / BLOCK;
  mvrenderer_softras_kernel<<<blocks, BLOCK, 0, stream>>>(feat, dists, zbuf, p2f,
                                                          out, npix);
}